// DGCNNView_45938970198559
// MI455X (gfx1250) — compile-verified
//
#include <hip/hip_runtime.h>

#define DIMC 512
#define NPTS 1024
#define BATCH 4
#define KNN 20
#define LSLOPE 0.2f
// 1/sqrt(1 + 1e-5)
#define BN_RS 0.9999950000374997f

typedef __attribute__((ext_vector_type(2))) float v2f;
typedef __attribute__((ext_vector_type(8))) float v8f;

static __device__ __forceinline__ v8f wmma_f32(v2f a, v2f b, v8f c) {
  // D = A(16x4 f32) * B(4x16 f32) + C(16x16 f32)   -> v_wmma_f32_16x16x4_f32
  return __builtin_amdgcn_wmma_f32_16x16x4_f32(false, a, false, b, (short)0, c,
                                               false, false);
}

static __device__ __forceinline__ float neg_inf() { return -__builtin_inff(); }

// ---------------------------------------------------------------------------
// xx[b*N+n] = sum_c x[b,c,n]^2
// ---------------------------------------------------------------------------
__global__ void norms_kernel(const float* __restrict__ x, float* __restrict__ xx) {
  int bn = blockIdx.x * blockDim.x + threadIdx.x;
  if (bn >= BATCH * NPTS) return;
  int b = bn >> 10, n = bn & (NPTS - 1);
  const float* xp = x + (size_t)b * DIMC * NPTS + n;
  float s = 0.f;
  for (int c = 0; c < DIMC; ++c) {
    float v = xp[(size_t)c * NPTS];
    s += v * v;
  }
  xx[bn] = s;
}

// ---------------------------------------------------------------------------
// pd[b,i,j] = 2*(x_i . x_j) - ||x_i||^2 - ||x_j||^2
// One wave per 32x64 block: 2 row-tiles x 4 col-tiles, 8 accumulators.
// Register blocking: 12 loads per 8 WMMA (vs 4 per 1 unblocked).
// ---------------------------------------------------------------------------
#define GR_RT 2
#define GR_CT 4
__global__ void gram_kernel(const float* __restrict__ x,
                            const float* __restrict__ xx,
                            float* __restrict__ pd) {
  const int lane = threadIdx.x & 31;
  const int m0 = blockIdx.x * (16 * GR_RT);  // row block (i)
  const int n0 = blockIdx.y * (16 * GR_CT);  // col block (j)
  const int b  = blockIdx.z;
  const float* xb = x + (size_t)b * DIMC * NPTS;
  const int l  = lane & 15;
  const int kh = lane >> 4;                  // 0/1 -> K pair {0,1} or {2,3}
  v8f acc[GR_RT][GR_CT] = {};
  for (int k0 = 0; k0 < DIMC; k0 += 4) {
    const int k = k0 + 2 * kh;
    const float* r0 = xb + (size_t)k * NPTS;
    const float* r1 = xb + (size_t)(k + 1) * NPTS;
    v2f av[GR_RT], bv[GR_CT];
#pragma unroll
    for (int i = 0; i < GR_RT; ++i) {        // A[m][k], coalesced over m
      av[i].x = r0[m0 + 16 * i + l];
      av[i].y = r1[m0 + 16 * i + l];
    }
#pragma unroll
    for (int j = 0; j < GR_CT; ++j) {        // B[k][n], coalesced over n
      bv[j].x = r0[n0 + 16 * j + l];
      bv[j].y = r1[n0 + 16 * j + l];
    }
#pragma unroll
    for (int i = 0; i < GR_RT; ++i)
#pragma unroll
      for (int j = 0; j < GR_CT; ++j)
        acc[i][j] = wmma_f32(av[i], bv[j], acc[i][j]);
  }
#pragma unroll
  for (int i = 0; i < GR_RT; ++i) {
#pragma unroll
    for (int j = 0; j < GR_CT; ++j) {
      const float xxj = xx[b * NPTS + n0 + 16 * j + l];
      float* pdb = pd + ((size_t)b * NPTS + m0 + 16 * i) * NPTS + n0 + 16 * j;
#pragma unroll
      for (int r = 0; r < 8; ++r) {
        const int M = r + 8 * kh;            // C layout: lanes16-31 -> M=r+8
        const float v =
            2.0f * acc[i][j][r] - xx[b * NPTS + m0 + 16 * i + M] - xxj;
        pdb[(size_t)M * NPTS + l] = v;       // coalesced store
      }
    }
  }
}

// ---------------------------------------------------------------------------
// top-K (K=20) per pd row; one wave32 per row, 32 elems/lane in registers
// ---------------------------------------------------------------------------
__global__ void topk_kernel(const float* __restrict__ pd, int* __restrict__ idx) {
  const int lane = threadIdx.x & 31;
  const int row  = blockIdx.x * (blockDim.x >> 5) + (threadIdx.x >> 5);
  if (row >= BATCH * NPTS) return;
  const float* r = pd + (size_t)row * NPTS;
  float vals[NPTS / 32];
#pragma unroll
  for (int i = 0; i < NPTS / 32; ++i) vals[i] = r[i * 32 + lane];
  unsigned taken = 0u;
  int* out = idx + row * KNN;
  for (int s = 0; s < KNN; ++s) {
    float best = neg_inf();
    int bi = NPTS;                 // sentinel larger than any index
#pragma unroll
    for (int i = 0; i < NPTS / 32; ++i) {
      if (!((taken >> i) & 1u)) {
        const float v = vals[i];
        const int j = i * 32 + lane;
        if (v > best || (v == best && j < bi)) { best = v; bi = j; }
      }
    }
    // wave32 arg-max reduction, ties -> lowest index (matches lax.top_k)
    for (int off = 16; off; off >>= 1) {
      const float ov = __shfl_xor(best, off);
      const int oi = __shfl_xor(bi, off);
      if (ov > best || (ov == best && oi < bi)) { best = ov; bi = oi; }
    }
    if (lane == (bi & 31)) taken |= 1u << (bi >> 5);
    if (lane == 0) out[s] = bi;
  }
}

// ---------------------------------------------------------------------------
// Pt[b,n,o] = (W_a x)[o,n] ; Rt[b,n,o] = ((W_b - W_a) x)[o,n]
// computed as (x^T W^T): M=n rows, N=o cols -> coalesced x loads & Pt/Rt stores
// One wave per 32x32 block: 2 n-tiles x 2 o-tiles, P and Q share x loads.
// ---------------------------------------------------------------------------
__global__ void proj_kernel(const float* __restrict__ x,
                            const float* __restrict__ w,  // (DIMC, 2*DIMC) row-major
                            float* __restrict__ Pt, float* __restrict__ Rt) {
  const int lane = threadIdx.x & 31;
  const int n0 = blockIdx.x * 32;
  const int o0 = blockIdx.y * 32;
  const int b  = blockIdx.z;
  const float* xb = x + (size_t)b * DIMC * NPTS;
  const int l  = lane & 15;
  const int kh = lane >> 4;
  v8f aP[2][2] = {}, aQ[2][2] = {};   // [n-tile][o-tile]
  for (int k0 = 0; k0 < DIMC; k0 += 4) {
    const int k = k0 + 2 * kh;
    v2f av[2], bP[2], bQ[2];
#pragma unroll
    for (int i = 0; i < 2; ++i) {     // A[m=n][k=c], coalesced
      av[i].x = xb[(size_t)k * NPTS + n0 + 16 * i + l];
      av[i].y = xb[(size_t)(k + 1) * NPTS + n0 + 16 * i + l];
    }
#pragma unroll
    for (int j = 0; j < 2; ++j) {     // B[k][n=o] = W_a^T / W_b^T
      const float* wr = w + (size_t)(o0 + 16 * j + l) * (2 * DIMC);
      bP[j].x = wr[k];          bP[j].y = wr[k + 1];
      bQ[j].x = wr[DIMC + k];   bQ[j].y = wr[DIMC + k + 1];
    }
#pragma unroll
    for (int i = 0; i < 2; ++i)
#pragma unroll
      for (int j = 0; j < 2; ++j) {
        aP[i][j] = wmma_f32(av[i], bP[j], aP[i][j]);
        aQ[i][j] = wmma_f32(av[i], bQ[j], aQ[i][j]);
      }
  }
#pragma unroll
  for (int i = 0; i < 2; ++i) {
#pragma unroll
    for (int r = 0; r < 8; ++r) {
      const int n = n0 + 16 * i + r + 8 * kh;   // D rows = n
      float* pt = Pt + ((size_t)b * NPTS + n) * DIMC + o0;
      float* rt = Rt + ((size_t)b * NPTS + n) * DIMC + o0;
#pragma unroll
      for (int j = 0; j < 2; ++j) {
        const float p = aP[i][j][r];
        pt[16 * j + l] = p;                     // coalesced
        rt[16 * j + l] = aQ[i][j][r] - p;       // R = Q - P
      }
    }
  }
}

// ---------------------------------------------------------------------------
// xout[b,o,n] = max_k act(scale[o]*(Pt[b,idx[n,k],o] + Rt[b,n,o]) + bias[o])
// block = 512 threads (o), grid = (N, B); gathers are 2KB contiguous rows
// ---------------------------------------------------------------------------
__global__ void edge_kernel(const float* __restrict__ Pt,
                            const float* __restrict__ Rt,
                            const int* __restrict__ idx,
                            const float* __restrict__ g,
                            const float* __restrict__ bb,
                            float* __restrict__ xout) {
  const int o = threadIdx.x;
  const int n = blockIdx.x;
  const int b = blockIdx.y;
  const float scale = g[o] * BN_RS;
  const float bias  = bb[o];
  const float* ptb = Pt + (size_t)b * NPTS * DIMC;
  const float rc = Rt[((size_t)b * NPTS + n) * DIMC + o];
  const int* id = idx + (b * NPTS + n) * KNN;
  float best = neg_inf();
#pragma unroll 4
  for (int k = 0; k < KNN; ++k) {
    float y = ptb[(size_t)id[k] * DIMC + o] + rc;
    y = scale * y + bias;
    y = (y >= 0.f) ? y : LSLOPE * y;
    best = fmaxf(best, y);
  }
  xout[((size_t)b * DIMC + o) * NPTS + n] = best;
}

// ---------------------------------------------------------------------------
// y[b,o,n] = act(s5[o] * (w5 @ [x1;x2;x3])[b,o,n] + b5[o])
// M=o rows from w5, N=n cols from x; 2x2 register blocking.
// ---------------------------------------------------------------------------
__global__ void final_proj_kernel(const float* __restrict__ x1,
                                  const float* __restrict__ x2,
                                  const float* __restrict__ x3,
                                  const float* __restrict__ w5, // (DIMC, 3*DIMC)
                                  const float* __restrict__ g5,
                                  const float* __restrict__ b5,
                                  float* __restrict__ y) {
  const int lane = threadIdx.x & 31;
  const int n0 = blockIdx.x * 32;
  const int o0 = blockIdx.y * 32;
  const int b  = blockIdx.z;
  const int l  = lane & 15;
  const int kh = lane >> 4;
  v8f acc[2][2] = {};   // [o-tile][n-tile]
  for (int k0 = 0; k0 < 3 * DIMC; k0 += 4) {
    const int k = k0 + 2 * kh;
    const float* xs = (k0 < DIMC) ? x1 : ((k0 < 2 * DIMC) ? x2 : x3);
    const int cc = (k0 & (DIMC - 1)) + 2 * kh;
    const float* xb = xs + (size_t)b * DIMC * NPTS;
    v2f av[2], bv[2];
#pragma unroll
    for (int j = 0; j < 2; ++j) {     // A[m=o][k=c]
      av[j].x = w5[(size_t)(o0 + 16 * j + l) * (3 * DIMC) + k];
      av[j].y = w5[(size_t)(o0 + 16 * j + l) * (3 * DIMC) + k + 1];
    }
#pragma unroll
    for (int i = 0; i < 2; ++i) {     // B[k][n], coalesced
      bv[i].x = xb[(size_t)cc * NPTS + n0 + 16 * i + l];
      bv[i].y = xb[(size_t)(cc + 1) * NPTS + n0 + 16 * i + l];
    }
#pragma unroll
    for (int j = 0; j < 2; ++j)
#pragma unroll
      for (int i = 0; i < 2; ++i)
        acc[j][i] = wmma_f32(av[j], bv[i], acc[j][i]);
  }
#pragma unroll
  for (int j = 0; j < 2; ++j) {
#pragma unroll
    for (int r = 0; r < 8; ++r) {
      const int o = o0 + 16 * j + r + 8 * kh;
      const float s = g5[o] * BN_RS;
#pragma unroll
      for (int i = 0; i < 2; ++i) {
        float v = s * acc[j][i][r] + b5[o];
        v = (v >= 0.f) ? v : LSLOPE * v;
        y[((size_t)b * DIMC + o) * NPTS + n0 + 16 * i + l] = v;  // coalesced
      }
    }
  }
}

// ---------------------------------------------------------------------------
// out[b*DIMC+o] = max_n y[b,o,n]   (one wave32 per row)
// ---------------------------------------------------------------------------
__global__ void reduce_max_kernel(const float* __restrict__ y,
                                  float* __restrict__ out) {
  const int lane = threadIdx.x & 31;
  const int row  = blockIdx.x * (blockDim.x >> 5) + (threadIdx.x >> 5);
  if (row >= BATCH * DIMC) return;
  const float* r = y + (size_t)row * NPTS;
  float m = neg_inf();
  for (int i = lane; i < NPTS; i += 32) m = fmaxf(m, r[i]);
  for (int off = 16; off; off >>= 1) m = fmaxf(m, __shfl_xor(m, off));
  if (lane == 0) out[row] = m;
}

// ---------------------------------------------------------------------------
extern "C" void kernel_launch(void* const* d_in, const int* in_sizes, int n_in,
                              void* d_out, int out_size, void* d_ws, size_t ws_size,
                              hipStream_t stream) {
  const float* x  = (const float*)d_in[0];
  // d_in[1] = pos (unused by the reference computation)
  const float* wl[3] = {(const float*)d_in[2], (const float*)d_in[3], (const float*)d_in[4]};
  const float* w5 = (const float*)d_in[5];
  const float* gl[3] = {(const float*)d_in[6], (const float*)d_in[8], (const float*)d_in[10]};
  const float* bl[3] = {(const float*)d_in[7], (const float*)d_in[9], (const float*)d_in[11]};
  const float* g5 = (const float*)d_in[12];
  const float* b5 = (const float*)d_in[13];
  float* out = (float*)d_out;

  // workspace carving
  const size_t BN  = (size_t)BATCH * NPTS;            // 4096
  const size_t BDN = (size_t)BATCH * DIMC * NPTS;     // 2M floats
  float* pd = (float*)d_ws;                           // B*N*N floats (reused as y)
  float* xx = pd + BN * NPTS;
  int*   id = (int*)(xx + BN);
  float* Pt = (float*)(id + BN * KNN);
  float* Rt = Pt + BDN;
  float* xs[3];
  xs[0] = Rt + BDN;
  xs[1] = xs[0] + BDN;
  xs[2] = xs[1] + BDN;
  float* yb = pd;  // reuse pd for the final activation buffer

  const dim3 gram_grid(NPTS / (16 * GR_RT), NPTS / (16 * GR_CT), BATCH);
  const dim3 proj_grid(NPTS / 32, DIMC / 32, BATCH);
  const dim3 edge_grid(NPTS, BATCH);

  const float* cur = x;
  for (int layer = 0; layer < 3; ++layer) {
    norms_kernel<<<(BATCH * NPTS + 255) / 256, 256, 0, stream>>>(cur, xx);
    gram_kernel<<<gram_grid, 32, 0, stream>>>(cur, xx, pd);
    topk_kernel<<<(BATCH * NPTS) / 8, 256, 0, stream>>>(pd, id);
    proj_kernel<<<proj_grid, 32, 0, stream>>>(cur, wl[layer], Pt, Rt);
    edge_kernel<<<edge_grid, DIMC, 0, stream>>>(Pt, Rt, id, gl[layer], bl[layer], xs[layer]);
    cur = xs[layer];
  }
  final_proj_kernel<<<proj_grid, 32, 0, stream>>>(xs[0], xs[1], xs[2], w5, g5, b5, yb);
  reduce_max_kernel<<<(BATCH * DIMC) / 8, 256, 0, stream>>>(yb, out);
}